// CachedCausalSelfAttention_1133871366573
// MI455X (gfx1250) — compile-verified
//
#include <hip/hip_runtime.h>

typedef __attribute__((ext_vector_type(2))) float v2f;
typedef __attribute__((ext_vector_type(8))) float v8f;

static constexpr int Bb = 16, Hh = 32, HD = 128, Dd = 4096, SP = 4095, Ss = 4096, E3 = 12288;
#define ATTN_SCALE 0.08838834764831845f /* 1/sqrt(128) */

// D = A(16x4 f32) * B(4x16 f32) + C(16x16 f32)
__device__ __forceinline__ v8f wmma4(v2f a, v2f b, v8f c) {
  return __builtin_amdgcn_wmma_f32_16x16x4_f32(
      /*neg_a=*/false, a, /*neg_b=*/false, b,
      /*c_mod=*/(short)0, c, /*reuse_a=*/false, /*reuse_b=*/false);
}

// ---------------------------------------------------------------------------
// Kernel 1: fused QKV projection  qkv[16][12288] = x[16][4096] @ Wqkv^T + b
// One wave = one 16x16 output tile (M = batch, N = output col), K-loop 4096.
// ---------------------------------------------------------------------------
__global__ void qkv_proj_kernel(const float* __restrict__ x,
                                const float* __restrict__ W,     // (12288,4096) row-major
                                const float* __restrict__ bias,  // (12288)
                                float* __restrict__ qkv)         // (16,12288)
{
  const int lane = threadIdx.x & 31;
  const int wave = threadIdx.x >> 5;
  const int n0   = (blockIdx.x * 8 + wave) * 16;
  const int mn   = lane & 15;            // both M-row of A and N-col of B
  const int koff = (lane >> 4) * 2;      // K sub-offset per half-wave

  const float* arow = x + (size_t)mn * Dd;            // A[m][k] = x[m][k]
  const float* brow = W + (size_t)(n0 + mn) * Dd;     // B[k][n] = W[n0+n][k]
  v8f acc = {};
#pragma unroll 8
  for (int k0 = 0; k0 < Dd; k0 += 4) {
    v2f a  = *(const v2f*)(arow + k0 + koff);
    v2f bm = *(const v2f*)(brow + k0 + koff);
    acc = wmma4(a, bm, acc);
  }
  const float bv = bias[n0 + mn];
#pragma unroll
  for (int i = 0; i < 8; ++i) {
    int row = i + (lane >> 4) * 8;       // C/D layout: VGPR i -> M = i + 8*(lane/16)
    qkv[(size_t)row * E3 + n0 + mn] = acc[i] + bv;
  }
}

// ---------------------------------------------------------------------------
// Kernel 2: single-query attention per (b,h).  256 threads / block.
//   Phase B: scores via WMMA, A = K-cache rows (M), B = q replicated over N.
//   Phase C: softmax in LDS.
//   Phase D: out = sum_s w[s] * V[s][:], coalesced VALU streaming of V.
// ---------------------------------------------------------------------------
__global__ void attn_kernel(const float* __restrict__ kc,   // (16,32,4095,128)
                            const float* __restrict__ vc,   // (16,32,4095,128)
                            const float* __restrict__ qkv,  // (16,12288)
                            float* __restrict__ attn)       // (16,4096)
{
  __shared__ float s_q[HD];
  __shared__ float s_sc[Ss];
  __shared__ float s_red[256];

  const int tid  = threadIdx.x;
  const int lane = tid & 31;
  const int wave = tid >> 5;
  const int b    = blockIdx.x >> 5;
  const int h    = blockIdx.x & 31;

  const float* qp   = qkv + (size_t)b * E3 + h * HD;
  const float* knew = qkv + (size_t)b * E3 + Dd + h * HD;
  const float* vnew = qkv + (size_t)b * E3 + 2 * Dd + h * HD;
  const float* kbh  = kc + (size_t)(b * Hh + h) * SP * HD;
  const float* vbh  = vc + (size_t)(b * Hh + h) * SP * HD;

  if (tid < HD) s_q[tid] = qp[tid];
  __syncthreads();

  // ---- scores: each wave covers 512 rows = 32 tiles of 16 rows
  const int koff = (lane >> 4) * 2;
  for (int t = 0; t < 32; ++t) {
    const int s0   = wave * 512 + t * 16;
    const int srow = s0 + (lane & 15);
    // last logical row (s==4095) is the freshly projected k
    const float* krow = (srow < SP) ? (kbh + (size_t)srow * HD) : knew;
    v8f acc = {};
#pragma unroll
    for (int k0 = 0; k0 < HD; k0 += 4) {
      v2f a  = *(const v2f*)(krow + k0 + koff);
      v2f qb = *(const v2f*)(&s_q[k0 + koff]);   // B[k][n] = q[k] for all n
      acc = wmma4(a, qb, acc);
    }
    if ((lane & 15) == 0) {                      // n==0 lanes own unique rows
      const int base = s0 + (lane >> 4) * 8;
#pragma unroll
      for (int i = 0; i < 8; ++i) s_sc[base + i] = acc[i] * ATTN_SCALE;
    }
  }
  __syncthreads();

  // ---- block max
  float lmax = -3.0e38f;
  for (int i = tid; i < Ss; i += 256) lmax = fmaxf(lmax, s_sc[i]);
  s_red[tid] = lmax;
  __syncthreads();
  for (int off = 128; off > 0; off >>= 1) {
    if (tid < off) s_red[tid] = fmaxf(s_red[tid], s_red[tid + off]);
    __syncthreads();
  }
  const float gmax = s_red[0];
  __syncthreads();

  // ---- exp + block sum
  float lsum = 0.f;
  for (int i = tid; i < Ss; i += 256) {
    float e = __expf(s_sc[i] - gmax);
    s_sc[i] = e;
    lsum += e;
  }
  s_red[tid] = lsum;
  __syncthreads();
  for (int off = 128; off > 0; off >>= 1) {
    if (tid < off) s_red[tid] += s_red[tid + off];
    __syncthreads();
  }
  const float inv = 1.0f / s_red[0];
  __syncthreads();

  // ---- out[d] = inv * sum_s e[s] * V[s][d];  128 threads span one V row
  const int d    = tid & (HD - 1);
  const int half = tid >> 7;                 // two s-halves of 2048
  const int sBeg = half * 2048;
  const int sEnd = sBeg + 2048;
  const int sStop = (sEnd > SP) ? SP : sEnd;
  float acc = 0.f;
#pragma unroll 4
  for (int s = sBeg; s < sStop; ++s)
    acc = fmaf(s_sc[s], vbh[(size_t)s * HD + d], acc);
  if (sEnd == Ss) acc = fmaf(s_sc[SP], vnew[d], acc);

  s_red[tid] = acc;
  __syncthreads();
  if (tid < HD)
    attn[(size_t)b * Dd + h * HD + tid] = (s_red[tid] + s_red[tid + HD]) * inv;
}

// ---------------------------------------------------------------------------
// Kernel 3: output projection  out[16][4096] = attn[16][4096] @ Wout^T + b
// ---------------------------------------------------------------------------
__global__ void out_proj_kernel(const float* __restrict__ attn,
                                const float* __restrict__ W,     // (4096,4096)
                                const float* __restrict__ bias,  // (4096)
                                float* __restrict__ out)         // (16,4096)
{
  const int lane = threadIdx.x & 31;
  const int wave = threadIdx.x >> 5;
  const int n0   = (blockIdx.x * 8 + wave) * 16;
  const int mn   = lane & 15;
  const int koff = (lane >> 4) * 2;

  const float* arow = attn + (size_t)mn * Dd;
  const float* brow = W + (size_t)(n0 + mn) * Dd;
  v8f acc = {};
#pragma unroll 8
  for (int k0 = 0; k0 < Dd; k0 += 4) {
    v2f a  = *(const v2f*)(arow + k0 + koff);
    v2f bm = *(const v2f*)(brow + k0 + koff);
    acc = wmma4(a, bm, acc);
  }
  const float bv = bias[n0 + mn];
#pragma unroll
  for (int i = 0; i < 8; ++i)
    out[(size_t)(i + (lane >> 4) * 8) * Dd + n0 + mn] = acc[i] + bv;
}

// ---------------------------------------------------------------------------
extern "C" void kernel_launch(void* const* d_in, const int* in_sizes, int n_in,
                              void* d_out, int out_size, void* d_ws, size_t ws_size,
                              hipStream_t stream) {
  const float* x    = (const float*)d_in[0];   // (16,1,4096)
  const float* kc   = (const float*)d_in[1];   // (16,32,4095,128)
  const float* vc   = (const float*)d_in[2];   // (16,32,4095,128)
  const float* Wqkv = (const float*)d_in[3];   // (12288,4096)
  const float* bqkv = (const float*)d_in[4];   // (12288)
  const float* Wout = (const float*)d_in[5];   // (4096,4096)
  const float* bout = (const float*)d_in[6];   // (4096)
  float* out = (float*)d_out;                  // (16,1,4096)

  float* qkv  = (float*)d_ws;                  // 16*12288 floats = 768 KB
  float* attn = qkv + (size_t)Bb * E3;         // 16*4096  floats = 256 KB

  qkv_proj_kernel<<<E3 / 128, 256, 0, stream>>>(x, Wqkv, bqkv, qkv);
  attn_kernel<<<Bb * Hh, 256, 0, stream>>>(kc, vc, qkv, attn);
  out_proj_kernel<<<Dd / 128, 256, 0, stream>>>(attn, Wout, bout, out);
}